// BatchPrototypeHead_79542794322400
// MI455X (gfx1250) — compile-verified
//
#include <hip/hip_runtime.h>
#include <hip/hip_bf16.h>

typedef float v2f __attribute__((ext_vector_type(2)));
typedef float v8f __attribute__((ext_vector_type(8)));

#define DM 256          // D_MODEL
#define NC 100          // N_CLASSES
#define NCP 112         // padded to 7 * 16 col tiles
#define HSTR 260        // LDS row stride (floats) to dodge bank conflicts
#define RCHUNKS 64
#define SROWS 65536
#define QROWS 8192
#define ROWS_PER_CHUNK (SROWS / RCHUNKS)

static __device__ inline v8f wmma_f32_k4(v2f a, v2f b, v8f c) {
  // D = A(16x4) * B(4x16) + C, exact fp32
  return __builtin_amdgcn_wmma_f32_16x16x4_f32(false, a, false, b, (short)0, c,
                                               false, false);
}

static __device__ inline v8f v8f_zero() {
  v8f z = {0.f, 0.f, 0.f, 0.f, 0.f, 0.f, 0.f, 0.f};
  return z;
}

// -------------------------------------------------------------------------
// Y = relu(X @ W1^T + b1) @ W2^T + b2      X:(nrows,256) -> Y:(nrows,256)
// block: 256 threads (8 waves), 32 rows per block.
// wave w: row tiles {0,1}, col tiles {2w, 2w+1}  -> W read exactly once/block
// -------------------------------------------------------------------------
__global__ __launch_bounds__(256) void projector_kernel(
    const float* __restrict__ X, const float* __restrict__ W1,
    const float* __restrict__ b1, const float* __restrict__ W2,
    const float* __restrict__ b2, float* __restrict__ Y) {
  __shared__ float sH[32 * HSTR];

  const int tid = threadIdx.x;
  const int wave = tid >> 5;
  const int lane = tid & 31;
  const int lr = lane & 15;
  const int kh = lane >> 4;         // which half of the wave
  const long rowbase = (long)blockIdx.x * 32;
  const int ct0 = wave * 2;

  const float* Xb = X + rowbase * DM;

  // ---- layer 1: A from global X, B from global W1 ----
  v8f acc[2][2];
  acc[0][0] = v8f_zero(); acc[0][1] = v8f_zero();
  acc[1][0] = v8f_zero(); acc[1][1] = v8f_zero();

  for (int kb = 0; kb < DM; kb += 4) {
    const int ko = kb + 2 * kh;
    v2f a0 = *(const v2f*)(Xb + (long)lr * DM + ko);
    v2f a1 = *(const v2f*)(Xb + (long)(16 + lr) * DM + ko);
#pragma unroll
    for (int ct = 0; ct < 2; ++ct) {
      const int n = (ct0 + ct) * 16 + lr;
      v2f b = *(const v2f*)(W1 + (long)n * DM + ko);
      acc[0][ct] = wmma_f32_k4(a0, b, acc[0][ct]);
      acc[1][ct] = wmma_f32_k4(a1, b, acc[1][ct]);
    }
  }

  // bias + relu -> LDS
#pragma unroll
  for (int ct = 0; ct < 2; ++ct) {
    const int n = (ct0 + ct) * 16 + lr;
    const float bias = b1[n];
#pragma unroll
    for (int rt = 0; rt < 2; ++rt) {
#pragma unroll
      for (int v = 0; v < 8; ++v) {
        const int row = rt * 16 + v + 8 * kh;
        float val = acc[rt][ct][v] + bias;
        sH[row * HSTR + n] = val > 0.f ? val : 0.f;
      }
    }
  }
  __syncthreads();

  // ---- layer 2: A from LDS sH, B from global W2 ----
  acc[0][0] = v8f_zero(); acc[0][1] = v8f_zero();
  acc[1][0] = v8f_zero(); acc[1][1] = v8f_zero();

  for (int kb = 0; kb < DM; kb += 4) {
    const int ko = kb + 2 * kh;
    v2f a0 = *(const v2f*)(sH + lr * HSTR + ko);
    v2f a1 = *(const v2f*)(sH + (16 + lr) * HSTR + ko);
#pragma unroll
    for (int ct = 0; ct < 2; ++ct) {
      const int n = (ct0 + ct) * 16 + lr;
      v2f b = *(const v2f*)(W2 + (long)n * DM + ko);
      acc[0][ct] = wmma_f32_k4(a0, b, acc[0][ct]);
      acc[1][ct] = wmma_f32_k4(a1, b, acc[1][ct]);
    }
  }

  // bias -> global Y
#pragma unroll
  for (int ct = 0; ct < 2; ++ct) {
    const int n = (ct0 + ct) * 16 + lr;
    const float bias = b2[n];
#pragma unroll
    for (int rt = 0; rt < 2; ++rt) {
#pragma unroll
      for (int v = 0; v < 8; ++v) {
        const long row = rowbase + rt * 16 + v + 8 * kh;
        Y[row * DM + n] = acc[rt][ct][v] + bias;
      }
    }
  }
}

// -------------------------------------------------------------------------
// Deterministic segment partial sums:
// grid (64 row chunks, 4 col slices), block 64 threads.
// Thread t exclusively owns column cs*64+t of a 100x64 LDS accumulator,
// accumulates rows in fixed ascending order -> bitwise deterministic.
// -------------------------------------------------------------------------
__global__ __launch_bounds__(64) void proto_partial_kernel(
    const float* __restrict__ Ys, const int* __restrict__ labels,
    float* __restrict__ psum, float* __restrict__ pcnt) {
  __shared__ float acc[NC * 64];
  const int t = threadIdx.x;
  const int chunk = blockIdx.x;
  const int cs = blockIdx.y;
  const int col = cs * 64 + t;
  const long base = (long)chunk * ROWS_PER_CHUNK;

  for (int c = 0; c < NC; ++c) acc[c * 64 + t] = 0.f;

  for (int r = 0; r < ROWS_PER_CHUNK; ++r) {
    const int lab = labels[base + r];   // uniform across the wave -> scalar
    acc[lab * 64 + t] += Ys[(base + r) * DM + col];
  }

  for (int c = 0; c < NC; ++c)
    psum[((long)chunk * NC + c) * DM + col] = acc[c * 64 + t];

  // exact (integer-valued) counts, deterministic, done once per chunk
  if (cs == 0) {
    for (int c = t; c < NC; c += 64) {
      int count = 0;
      for (int r = 0; r < ROWS_PER_CHUNK; ++r)
        count += (labels[base + r] == c) ? 1 : 0;
      pcnt[chunk * NC + c] = (float)count;
    }
  }
}

// -------------------------------------------------------------------------
// protos[c] = sum/count (0 if empty); pn[c] = |proto_c|^2.
// 112 blocks: classes 100..111 are zero padding for the distance GEMM.
// -------------------------------------------------------------------------
__global__ __launch_bounds__(256) void proto_finalize_kernel(
    const float* __restrict__ psum, const float* __restrict__ pcnt,
    float* __restrict__ protos, float* __restrict__ pn) {
  __shared__ float red[256];
  const int c = blockIdx.x;
  const int d = threadIdx.x;

  float p = 0.f;
  if (c < NC) {
    float s = 0.f;
    for (int ch = 0; ch < RCHUNKS; ++ch)
      s += psum[((long)ch * NC + c) * DM + d];
    float cnt = 0.f;
    for (int ch = 0; ch < RCHUNKS; ++ch) cnt += pcnt[ch * NC + c];
    p = (cnt > 0.f) ? (s / cnt) : 0.f;
  }
  protos[c * DM + d] = p;

  red[d] = p * p;
  __syncthreads();
  for (int s = 128; s > 0; s >>= 1) {
    if (d < s) red[d] += red[d + s];
    __syncthreads();
  }
  if (d == 0) pn[c] = red[0];
}

// -------------------------------------------------------------------------
// qn[row] = |Yq[row]|^2 ; one wave per row, fixed-order shuffle reduction.
// -------------------------------------------------------------------------
__global__ __launch_bounds__(256) void qnorm_kernel(
    const float* __restrict__ Yq, float* __restrict__ qn) {
  const int wave = threadIdx.x >> 5;
  const int lane = threadIdx.x & 31;
  const long row = (long)blockIdx.x * 8 + wave;
  const float* r = Yq + row * DM;
  float s = 0.f;
#pragma unroll
  for (int i = 0; i < 8; ++i) {
    float v = r[lane + 32 * i];
    s += v * v;
  }
  for (int off = 16; off > 0; off >>= 1) s += __shfl_xor(s, off, 32);
  if (lane == 0) qn[row] = s;
}

// -------------------------------------------------------------------------
// out[b][c] = 2*(q.p) - |q|^2 - |p|^2  via WMMA GEMM Yq(8192x256) x P^T(256x112)
// block: 256 threads / 8 waves; wave -> one 16-row tile x 7 col tiles.
// -------------------------------------------------------------------------
__global__ __launch_bounds__(256) void dist_kernel(
    const float* __restrict__ Yq, const float* __restrict__ protos,
    const float* __restrict__ qn, const float* __restrict__ pn,
    float* __restrict__ out) {
  const int wave = threadIdx.x >> 5;
  const int lane = threadIdx.x & 31;
  const int lr = lane & 15;
  const int kh = lane >> 4;
  const long rowtile = (long)blockIdx.x * 8 + wave;
  const float* A = Yq + rowtile * 16 * DM;

  v8f acc[7];
#pragma unroll
  for (int ct = 0; ct < 7; ++ct) acc[ct] = v8f_zero();

  for (int kb = 0; kb < DM; kb += 4) {
    const int ko = kb + 2 * kh;
    v2f a = *(const v2f*)(A + (long)lr * DM + ko);
#pragma unroll
    for (int ct = 0; ct < 7; ++ct) {
      const int n = ct * 16 + lr;
      v2f b = *(const v2f*)(protos + (long)n * DM + ko);
      acc[ct] = wmma_f32_k4(a, b, acc[ct]);
    }
  }

  float qv[8];
#pragma unroll
  for (int v = 0; v < 8; ++v) qv[v] = qn[rowtile * 16 + v + 8 * kh];

#pragma unroll
  for (int ct = 0; ct < 7; ++ct) {
    const int n = ct * 16 + lr;
    if (n < NC) {
      const float pv = pn[n];
#pragma unroll
      for (int v = 0; v < 8; ++v) {
        const long row = rowtile * 16 + v + 8 * kh;
        out[row * NC + n] = 2.f * acc[ct][v] - qv[v] - pv;
      }
    }
  }
}

// -------------------------------------------------------------------------
extern "C" void kernel_launch(void* const* d_in, const int* in_sizes, int n_in,
                              void* d_out, int out_size, void* d_ws,
                              size_t ws_size, hipStream_t stream) {
  (void)in_sizes; (void)n_in; (void)out_size; (void)ws_size;
  const float* query = (const float*)d_in[0];
  const float* sup   = (const float*)d_in[1];
  const int*   labels = (const int*)d_in[2];
  const float* W1 = (const float*)d_in[3];
  const float* b1 = (const float*)d_in[4];
  const float* W2 = (const float*)d_in[5];
  const float* b2 = (const float*)d_in[6];
  float* out = (float*)d_out;

  float* ws = (float*)d_ws;
  float* Yq = ws;                                   // 8192*256
  float* Ys = Yq + (size_t)QROWS * DM;              // 65536*256
  float* psum = Ys + (size_t)SROWS * DM;            // 64*100*256
  float* pcnt = psum + (size_t)RCHUNKS * NC * DM;   // 64*100
  float* protos = pcnt + (size_t)RCHUNKS * NC;      // 112*256 (padded)
  float* pn = protos + (size_t)NCP * DM;            // 112
  float* qn = pn + NCP;                             // 8192

  projector_kernel<<<QROWS / 32, 256, 0, stream>>>(query, W1, b1, W2, b2, Yq);
  projector_kernel<<<SROWS / 32, 256, 0, stream>>>(sup, W1, b1, W2, b2, Ys);

  dim3 gpp(RCHUNKS, 4);
  proto_partial_kernel<<<gpp, 64, 0, stream>>>(Ys, labels, psum, pcnt);
  proto_finalize_kernel<<<NCP, 256, 0, stream>>>(psum, pcnt, protos, pn);
  qnorm_kernel<<<QROWS / 8, 256, 0, stream>>>(Yq, qn);
  dist_kernel<<<(QROWS / 16) / 8, 256, 0, stream>>>(Yq, protos, qn, pn, out);
}